// CausalRankKAttention_22265110463163
// MI455X (gfx1250) — compile-verified
//
#include <hip/hip_runtime.h>
#include <math.h>

typedef __attribute__((ext_vector_type(2))) float v2f;
typedef __attribute__((ext_vector_type(8))) float v8f;

#define NB    2
#define LSEQ  2048
#define NHEAD 8
#define EDIM  64
#define DDIM  64
#define NHD   (NB * NHEAD)          // 16 (n,h) pairs
#define RSTR  (NHEAD * EDIM)        // 512: stride between consecutive l rows
#define TEMP  0.125f                // 1/sqrt(64)
#define DIAG  128
#define EPSF  1e-6f
#define NEGBIG (-3.0e38f)

// ---------------- WMMA helpers (exact fp32 path: V_WMMA_F32_16X16X4_F32) ----

__device__ __forceinline__ v8f wmma4(v2f a, v2f b, v8f c) {
  return __builtin_amdgcn_wmma_f32_16x16x4_f32(false, a, false, b, (short)0, c,
                                               false, false);
}

// A-fragment (16x4, M x K) from row-major M x K source; also serves as the
// B-fragment (4x16, K x N) when the source stores B^T row-major (N x K).
// ISA layout: lane m=t&15; VGPR0 holds K = 2*(t>=16)+0, VGPR1 holds K+1.
__device__ __forceinline__ v2f frag_rowK(const float* src, int stride, int k0,
                                         int lane) {
  int m = lane & 15;
  int h2 = (lane >> 4) << 1;
  const float* p = src + (size_t)m * stride + k0 + h2;
  v2f r; r.x = p[0]; r.y = p[1];
  return r;
}

// B-fragment (K x N) from a row-major K x N source (e.g. V tile, S state);
// also serves as the A-fragment of a transposed source.
__device__ __forceinline__ v2f frag_colN(const float* src, int stride, int k0,
                                         int n0, int lane) {
  int n = lane & 15;
  int h2 = (lane >> 4) << 1;
  const float* p = src + (size_t)(k0 + h2) * stride + n0 + n;
  v2f r; r.x = p[0]; r.y = p[stride];
  return r;
}

// Row-sum over the N dimension of a C/D fragment (within each 16-lane half).
__device__ __forceinline__ float rowred_sum(float v) {
  v += __shfl_xor(v, 1);
  v += __shfl_xor(v, 2);
  v += __shfl_xor(v, 4);
  v += __shfl_xor(v, 8);
  return v;
}

// Cross-lane merge of per-lane online-softmax states (m, z) over 16 columns.
__device__ __forceinline__ void merge_mz(float& m, float& z) {
#pragma unroll
  for (int msk = 1; msk <= 8; msk <<= 1) {
    float mo = __shfl_xor(m, msk);
    float zo = __shfl_xor(z, msk);
    float mn = fmaxf(m, mo);
    z = z * __expf(m - mn) + zo * __expf(mo - mn);  // exp(NEGBIG-NEGBIG)=1, z=0 ok
    m = mn;
  }
}

// ============ Pass A: full causal row softmax statistics (m, Z) =============
// Per-lane online max/sum over the lane's column subset (s == ncol mod 16);
// one cross-lane merge at the end — no per-tile shuffles.
__global__ __launch_bounds__(32)
void stats_kernel(const float* __restrict__ Q, const float* __restrict__ K,
                  const float* __restrict__ mask,
                  float* __restrict__ mbuf, float* __restrict__ zbuf) {
  const int lane = threadIdx.x;
  const int nh = blockIdx.x;              // 0..15
  const int n = nh >> 3, h = nh & 7;
  const int qt = blockIdx.y;              // query tile, l0 = 16*qt
  const int l0 = qt * 16;
  const int half = lane >> 4;
  const int ncol = lane & 15;

  const float* qbase = Q + ((size_t)(n * LSEQ + l0) * NHEAD + h) * EDIM;
  v2f qa[16];
#pragma unroll
  for (int kk = 0; kk < 16; ++kk) qa[kk] = frag_rowK(qbase, RSTR, kk * 4, lane);

  float mrow[8], zrow[8];
#pragma unroll
  for (int j = 0; j < 8; ++j) { mrow[j] = NEGBIG; zrow[j] = 0.f; }

  for (int st = 0; st <= qt; ++st) {
    const int s0 = st * 16;
    const float* kbase = K + ((size_t)(n * LSEQ + s0) * NHEAD + h) * EDIM;
    if (st < qt) {
      // prefetch next K tile: 16 rows x 256B (each lane covers 128B of a row)
      const char* pf = (const char*)(kbase + (size_t)16 * RSTR);
      __builtin_prefetch(pf + (size_t)ncol * (RSTR * 4) + half * 128, 0, 0);
    }
    // Batched B-fragment loads -> clause of 16 b64 loads, counted waits.
    v2f kb[16];
#pragma unroll
    for (int kk = 0; kk < 16; ++kk)
      kb[kk] = frag_rowK(kbase, RSTR, kk * 4, lane);
    // Two independent accumulation chains for WMMA pipelining.
    v8f acc0 = {}, acc1 = {};
#pragma unroll
    for (int kk = 0; kk < 8; ++kk) {
      acc0 = wmma4(qa[2 * kk + 0], kb[2 * kk + 0], acc0);
      acc1 = wmma4(qa[2 * kk + 1], kb[2 * kk + 1], acc1);
    }

    const int s = s0 + ncol;
    const float km = mask[n * LSEQ + s];
#pragma unroll
    for (int j = 0; j < 8; ++j) {
      const int l = l0 + j + 8 * half;
      const bool valid = (s <= l) && (km > 0.f);
      const float logit = valid ? (TEMP * (acc0[j] + acc1[j])) : NEGBIG;
      const float mn = fmaxf(mrow[j], logit);
      const float p = (logit > -1.0e37f) ? __expf(logit - mn) : 0.f;
      zrow[j] = zrow[j] * __expf(mrow[j] - mn) + p;   // NEGBIG-NEGBIG==0
      mrow[j] = mn;
    }
  }
#pragma unroll
  for (int j = 0; j < 8; ++j) merge_mz(mrow[j], zrow[j]);
  if (ncol == 0) {
#pragma unroll
    for (int j = 0; j < 8; ++j) {
      const int l = l0 + j + 8 * half;
      mbuf[nh * LSEQ + l] = mrow[j];
      zbuf[nh * LSEQ + l] = zrow[j];
    }
  }
}

// ============ Pass B: banded softmax numerator, writes b*SV/Z ===============
__global__ __launch_bounds__(32)
void band_kernel(const float* __restrict__ Q, const float* __restrict__ K,
                 const float* __restrict__ V, const float* __restrict__ mask,
                 const float* __restrict__ blend,
                 const float* __restrict__ mbuf, const float* __restrict__ zbuf,
                 float* __restrict__ out) {
  __shared__ float P[16 * 20];            // padded: conflict-free frag reads
  const int lane = threadIdx.x;
  const int nh = blockIdx.x;
  const int n = nh >> 3, h = nh & 7;
  const int qt = blockIdx.y;
  const int l0 = qt * 16;
  const int half = lane >> 4;
  const int ncol = lane & 15;

  const float* qbase = Q + ((size_t)(n * LSEQ + l0) * NHEAD + h) * EDIM;
  v2f qa[16];
#pragma unroll
  for (int kk = 0; kk < 16; ++kk) qa[kk] = frag_rowK(qbase, RSTR, kk * 4, lane);

  float mrow[8], zinv[8];
#pragma unroll
  for (int j = 0; j < 8; ++j) {
    const int l = l0 + j + 8 * half;
    mrow[j] = mbuf[nh * LSEQ + l];
    zinv[j] = 1.f / fmaxf(zbuf[nh * LSEQ + l], 1e-30f);
  }

  v8f o0 = {}, o1 = {}, o2 = {}, o3 = {};
  const int st_lo = (qt >= 8) ? (qt - 8) : 0;   // band half-width 128 = 8 tiles

  for (int st = st_lo; st <= qt; ++st) {
    const int s0 = st * 16;
    const float* kbase = K + ((size_t)(n * LSEQ + s0) * NHEAD + h) * EDIM;
    const float* vbase = V + ((size_t)(n * LSEQ + s0) * NHEAD + h) * EDIM;
    // Issue K and V fragment loads up front; V loads overlap exp/LDS work.
    v2f kb[16];
#pragma unroll
    for (int kk = 0; kk < 16; ++kk)
      kb[kk] = frag_rowK(kbase, RSTR, kk * 4, lane);
    v2f vb[16];
#pragma unroll
    for (int dt = 0; dt < 4; ++dt)
#pragma unroll
      for (int kk = 0; kk < 4; ++kk)
        vb[dt * 4 + kk] = frag_colN(vbase, RSTR, kk * 4, dt * 16, lane);

    v8f acc0 = {}, acc1 = {};
#pragma unroll
    for (int kk = 0; kk < 8; ++kk) {
      acc0 = wmma4(qa[2 * kk + 0], kb[2 * kk + 0], acc0);
      acc1 = wmma4(qa[2 * kk + 1], kb[2 * kk + 1], acc1);
    }

    const int s = s0 + ncol;
    const float km = mask[n * LSEQ + s];
    __syncthreads();                       // WAR on P vs previous tile
#pragma unroll
    for (int j = 0; j < 8; ++j) {
      const int l = l0 + j + 8 * half;
      const bool valid = (s <= l) && (s + DIAG >= l) && (km > 0.f);
      const float w =
          valid ? __expf(TEMP * (acc0[j] + acc1[j]) - mrow[j]) : 0.f;
      P[(j + 8 * half) * 20 + ncol] = w;   // C-layout -> LDS (row-major 16x16)
    }
    __syncthreads();
    v2f pa[4];
#pragma unroll
    for (int kk = 0; kk < 4; ++kk) pa[kk] = frag_rowK(P, 20, kk * 4, lane);
#pragma unroll
    for (int kk = 0; kk < 4; ++kk) {
      o0 = wmma4(pa[kk], vb[0 * 4 + kk], o0);
      o1 = wmma4(pa[kk], vb[1 * 4 + kk], o1);
      o2 = wmma4(pa[kk], vb[2 * 4 + kk], o2);
      o3 = wmma4(pa[kk], vb[3 * 4 + kk], o3);
    }
  }

  const float b = blend[0];
#pragma unroll
  for (int j = 0; j < 8; ++j) {
    const int l = l0 + j + 8 * half;
    float* ob = out + ((size_t)(n * LSEQ + l) * NHEAD + h) * DDIM + ncol;
    ob[0]  = b * o0[j] * zinv[j];
    ob[16] = b * o1[j] * zinv[j];
    ob[32] = b * o2[j] * zinv[j];
    ob[48] = b * o3[j] * zinv[j];
  }
}

// ============ Pass C: chunked causal linear attention, out += (1-b)*LV ======
__global__ __launch_bounds__(32)
void linear_kernel(const float* __restrict__ Q, const float* __restrict__ K,
                   const float* __restrict__ V, const float* __restrict__ mask,
                   const float* __restrict__ blend, float* __restrict__ out) {
  __shared__ float S[64 * 68];            // running E x D state (padded)
  __shared__ float Qc[16 * 68], Kc[16 * 68], Vc[16 * 68];
  __shared__ float P[16 * 20];
  __shared__ float ksum[64];
  __shared__ float zl[16];
  const int lane = threadIdx.x;
  const int nh = blockIdx.x;
  const int n = nh >> 3, h = nh & 7;
  const int half = lane >> 4;
  const int ncol = lane & 15;
  const float wlin = 1.f - blend[0];

  for (int i = lane; i < 64 * 68; i += 32) S[i] = 0.f;
  if (lane < 32) { ksum[lane] = 0.f; ksum[lane + 32] = 0.f; }
  __syncthreads();

  for (int c = 0; c < LSEQ / 16; ++c) {
    const int l0 = c * 16;
    // stage feature-mapped chunk (coalesced along e)
    for (int idx = lane; idx < 16 * 64; idx += 32) {
      const int r = idx >> 6, e = idx & 63;
      const size_t g = ((size_t)(n * LSEQ + l0 + r) * NHEAD + h) * EDIM + e;
      const float mk = mask[n * LSEQ + l0 + r];
      Qc[r * 68 + e] = tanhf(Q[g]) + 1.f;
      Kc[r * 68 + e] = (tanhf(K[g]) + 1.f) * mk;
      Vc[r * 68 + e] = V[g];
    }
    __syncthreads();

    // inter normalizer: z_inter[m] = Q1[m] . ksum (row m = ncol, halves split e)
    {
      float part = 0.f;
      const int e0 = half * 32;
#pragma unroll
      for (int i = 0; i < 32; ++i) part += Qc[ncol * 68 + e0 + i] * ksum[e0 + i];
      part += __shfl_xor(part, 16);
      if (half == 0) zl[ncol] = part;
    }
    __syncthreads();

    v2f qa[16];
#pragma unroll
    for (int kk = 0; kk < 16; ++kk) qa[kk] = frag_rowK(Qc, 68, kk * 4, lane);

    // intra A2 = Q1 @ K1^T (causal-masked, no exp — linear attention)
    v2f kb[16];
#pragma unroll
    for (int kk = 0; kk < 16; ++kk) kb[kk] = frag_rowK(Kc, 68, kk * 4, lane);
    v8f acc0 = {}, acc1 = {};
#pragma unroll
    for (int kk = 0; kk < 8; ++kk) {
      acc0 = wmma4(qa[2 * kk + 0], kb[2 * kk + 0], acc0);
      acc1 = wmma4(qa[2 * kk + 1], kb[2 * kk + 1], acc1);
    }

    float zfin[8];
#pragma unroll
    for (int j = 0; j < 8; ++j) {
      const int M = j + 8 * half;
      const float w = (ncol <= M) ? (acc0[j] + acc1[j]) : 0.f;
      P[M * 20 + ncol] = w;
      zfin[j] = zl[M] + rowred_sum(w) + EPSF;
    }
    __syncthreads();
    v2f pa[4];
#pragma unroll
    for (int kk = 0; kk < 4; ++kk) pa[kk] = frag_rowK(P, 20, kk * 4, lane);

    // LV = Q1 @ S  +  tril(A2) @ V ; blend-accumulate into out
#pragma unroll
    for (int dt = 0; dt < 4; ++dt) {
      v2f sb[16];
#pragma unroll
      for (int kk = 0; kk < 16; ++kk)
        sb[kk] = frag_colN(S, 68, kk * 4, dt * 16, lane);
      v8f oa = {}, ob = {};
#pragma unroll
      for (int kk = 0; kk < 8; ++kk) {
        oa = wmma4(qa[2 * kk + 0], sb[2 * kk + 0], oa);
        ob = wmma4(qa[2 * kk + 1], sb[2 * kk + 1], ob);
      }
#pragma unroll
      for (int kk = 0; kk < 4; ++kk)
        oa = wmma4(pa[kk], frag_colN(Vc, 68, kk * 4, dt * 16, lane), oa);
#pragma unroll
      for (int j = 0; j < 8; ++j) {
        const int l = l0 + j + 8 * half;
        const int d = dt * 16 + ncol;
        const size_t gi = ((size_t)(n * LSEQ + l) * NHEAD + h) * DDIM + d;
        out[gi] += wlin * (oa[j] + ob[j]) / zfin[j];
      }
    }
    __syncthreads();                       // S reads done before S update

    // S += K1^T @ V via WMMA; each C-fragment cell owned by a unique lane
#pragma unroll
    for (int et = 0; et < 4; ++et)
#pragma unroll
      for (int dt = 0; dt < 4; ++dt) {
        v8f ds = {};
#pragma unroll
        for (int kk = 0; kk < 4; ++kk)
          ds = wmma4(frag_colN(Kc, 68, kk * 4, et * 16, lane),   // (K1^T) A-frag
                     frag_colN(Vc, 68, kk * 4, dt * 16, lane), ds);
#pragma unroll
        for (int j = 0; j < 8; ++j) {
          const int e = et * 16 + j + 8 * half;
          const int d = dt * 16 + ncol;
          S[e * 68 + d] += ds[j];
        }
      }
    // ksum += colsum(K1)
    for (int e = lane; e < 64; e += 32) {
      float a = 0.f;
#pragma unroll
      for (int s = 0; s < 16; ++s) a += Kc[s * 68 + e];
      ksum[e] += a;
    }
    __syncthreads();
  }
}

// ============================ launch ========================================
extern "C" void kernel_launch(void* const* d_in, const int* in_sizes, int n_in,
                              void* d_out, int out_size, void* d_ws,
                              size_t ws_size, hipStream_t stream) {
  (void)in_sizes; (void)n_in; (void)out_size; (void)ws_size;
  const float* Q     = (const float*)d_in[0];
  const float* K     = (const float*)d_in[1];
  const float* V     = (const float*)d_in[2];
  const float* mask  = (const float*)d_in[3];
  const float* blend = (const float*)d_in[4];
  float* out  = (float*)d_out;
  float* mbuf = (float*)d_ws;              // [NHD*LSEQ] row max
  float* zbuf = mbuf + NHD * LSEQ;         // [NHD*LSEQ] row sum

  dim3 grid(NHD, LSEQ / 16);
  hipLaunchKernelGGL(stats_kernel,  grid, dim3(32), 0, stream, Q, K, mask,
                     mbuf, zbuf);
  hipLaunchKernelGGL(band_kernel,   grid, dim3(32), 0, stream, Q, K, V, mask,
                     blend, mbuf, zbuf, out);
  hipLaunchKernelGGL(linear_kernel, dim3(NHD), dim3(32), 0, stream, Q, K, V,
                     mask, blend, out);
}